// SymEqNet_2911987826902
// MI455X (gfx1250) — compile-verified
//
#include <hip/hip_runtime.h>
#include <hip/hip_bf16.h>

typedef __attribute__((ext_vector_type(2))) float v2f;
typedef __attribute__((ext_vector_type(8))) float v8f;

#define NN 500000
#define EE 8000000
#define DD 10
#define GG 5000

// ---------------------------------------------------------------- zero scratch
__global__ void zero_kernel(float* __restrict__ p, long long n) {
    long long i = (long long)blockIdx.x * blockDim.x + threadIdx.x;
    long long stride = (long long)gridDim.x * blockDim.x;
    for (; i < n; i += stride) p[i] = 0.0f;
}

// ------------------------------------------------- edge scatter: h += w * x[src]
__global__ void edge_scatter(const float* __restrict__ x,
                             const int* __restrict__ ei,
                             const float* __restrict__ ea,
                             float* __restrict__ h) {
    int e = blockIdx.x * blockDim.x + threadIdx.x;
    if (e >= EE) return;
    // edge arrays are streamed exactly once: non-temporal so x/h keep L2
    int src = __builtin_nontemporal_load(ei + e);
    int dst = __builtin_nontemporal_load(ei + EE + e);
    float w = __builtin_nontemporal_load(ea + e);
    const float2* xr = (const float2*)(x + (long long)src * DD);   // rows are 8B aligned
    float* hr = h + (long long)dst * DD;
    float2 v0 = xr[0], v1 = xr[1], v2 = xr[2], v3 = xr[3], v4 = xr[4];
    atomicAdd(&hr[0], v0.x * w); atomicAdd(&hr[1], v0.y * w);
    atomicAdd(&hr[2], v1.x * w); atomicAdd(&hr[3], v1.y * w);
    atomicAdd(&hr[4], v2.x * w); atomicAdd(&hr[5], v2.y * w);
    atomicAdd(&hr[6], v3.x * w); atomicAdd(&hr[7], v3.y * w);
    atomicAdd(&hr[8], v4.x * w); atomicAdd(&hr[9], v4.y * w);
}

// ------------------------------------------------- pool nodes into graph sums
__global__ void node_pool(const float* __restrict__ h,
                          const int* __restrict__ batch,
                          float* __restrict__ s,
                          float* __restrict__ cnt) {
    int i = blockIdx.x * blockDim.x + threadIdx.x;
    if (i >= NN) return;
    int b = batch[i];
    const float2* hr = (const float2*)(h + (long long)i * DD);
    float* sr = s + (long long)b * DD;
    float2 v0 = hr[0], v1 = hr[1], v2 = hr[2], v3 = hr[3], v4 = hr[4];
    atomicAdd(&sr[0], v0.x); atomicAdd(&sr[1], v0.y);
    atomicAdd(&sr[2], v1.x); atomicAdd(&sr[3], v1.y);
    atomicAdd(&sr[4], v2.x); atomicAdd(&sr[5], v2.y);
    atomicAdd(&sr[6], v3.x); atomicAdd(&sr[7], v3.y);
    atomicAdd(&sr[8], v4.x); atomicAdd(&sr[9], v4.y);
    atomicAdd(&cnt[b], 1.0f);
}

// ------------------------------------------------- g = s / max(cnt, 1)
__global__ void pool_div(const float* __restrict__ s,
                         const float* __restrict__ cnt,
                         float* __restrict__ g) {
    int idx = blockIdx.x * blockDim.x + threadIdx.x;
    if (idx >= GG * DD) return;
    float c = cnt[idx / DD];
    g[idx] = s[idx] / fmaxf(c, 1.0f);
}

// ------------------------------------------------- WMMA GEMM: Z = Y @ W^T + b, + column stats
// 256 threads = 8 waves = 8 x 16-row tiles per block.  Operands staged through LDS
// (zero-padded to K=12) so every WMMA source is an unconditional 8B-aligned ds_load_b64.
// V_WMMA_F32_16X16X4_F32 x3 per tile (K = 10 padded to 12).
// A layout (16x4 f32): lane&15 = M; VGPR0/1 = K {k0,k0+1} lanes 0-15, {k0+2,k0+3} lanes 16-31.
// B layout (4x16 f32): lane&15 = N; same K split across lane halves.
// C/D layout (16x16 f32): VGPR r -> M = r + 8*(lane>=16), N = lane&15.
__global__ __launch_bounds__(256) void gemm_wmma(const float* __restrict__ Y,
                                                 const float* __restrict__ W,
                                                 const float* __restrict__ bias,
                                                 float* __restrict__ Z,
                                                 float* __restrict__ colsum,
                                                 float* __restrict__ colsq,
                                                 int rows) {
    __shared__ float Wp[16 * 12];     // B operand, padded
    __shared__ float Yp[128 * 12];    // A operands for 8 tiles, padded

    const int tid = threadIdx.x;
    const int rowbase = blockIdx.x * 128;

    // stage W^T source (B[k][n] = W[n][k]) as Wp[n*12 + k], zero-padded
    for (int idx = tid; idx < 16 * 12; idx += 256) {
        int nn = idx / 12, kk = idx % 12;
        Wp[idx] = (nn < DD && kk < DD) ? W[nn * DD + kk] : 0.0f;
    }
    // stage Y block as Yp[r*12 + k], zero-padded rows/cols
    for (int idx = tid; idx < 128 * 12; idx += 256) {
        int r = idx / 12, kk = idx % 12;
        int row = rowbase + r;
        Yp[idx] = (row < rows && kk < DD) ? Y[row * DD + kk] : 0.0f;
    }
    __syncthreads();

    const int lane = tid & 31;
    const int wave = tid >> 5;
    const int tile = blockIdx.x * 8 + wave;
    const int ntiles = (rows + 15) >> 4;
    if (tile < ntiles) {                       // wave-uniform: EXEC all-1s for WMMA
        const int n    = lane & 15;            // output column (N) for B/C/D
        const int half = lane >> 4;
        const int arow = wave * 16 + (lane & 15);

        const float bn = (n < DD) ? bias[n] : 0.0f;
        v8f c;
#pragma unroll
        for (int r = 0; r < 8; ++r) c[r] = bn;   // bias folded into C operand

#pragma unroll
        for (int k0 = 0; k0 < 12; k0 += 4) {
            const int ka = k0 + half * 2;                       // even -> 8B aligned
            v2f a = *(const v2f*)&Yp[arow * 12 + ka];           // ds_load_b64
            v2f b = *(const v2f*)&Wp[n * 12 + ka];              // ds_load_b64
            c = __builtin_amdgcn_wmma_f32_16x16x4_f32(false, a, false, b,
                                                      (short)0, c, false, false);
        }

        // rows (=5000) is a multiple of 8, so each aligned 8-row half-tile is
        // either entirely valid or entirely invalid -> one uniform predicate,
        // unconditional stores inside (no per-row exec juggling).
        const int r0 = (tile << 4) + half * 8;
        if (n < DD && r0 < rows) {
            float sm = 0.0f, sq = 0.0f;
#pragma unroll
            for (int r = 0; r < 8; ++r) {
                const float v = c[r];
                Z[(r0 + r) * DD + n] = v;
                sm += v;
                sq += v * v;
            }
            atomicAdd(&colsum[n], sm);
            atomicAdd(&colsq[n], sq);
        }
    }
}

// ------------------------------------------------- batchnorm (+optional residual) + relu, in place
__global__ void bn_apply(float* __restrict__ Z,
                         const float* __restrict__ colsum,
                         const float* __restrict__ colsq,
                         const float* __restrict__ gamma,
                         const float* __restrict__ beta,
                         const float* __restrict__ res,
                         int rows, int has_res) {
    int idx = blockIdx.x * blockDim.x + threadIdx.x;
    if (idx >= rows * DD) return;
    int col = idx % DD;
    float invG = 1.0f / (float)rows;
    float mean = colsum[col] * invG;
    float var  = colsq[col] * invG - mean * mean;   // biased var, matches reference
    float inv  = rsqrtf(var + 1e-5f);
    float v = gamma[col] * (Z[idx] - mean) * inv + beta[col];
    if (has_res) v += res[idx];
    Z[idx] = fmaxf(v, 0.0f);
}

// ------------------------------------------------- final 10 -> 1 projection
__global__ void out_proj(const float* __restrict__ Y,
                         const float* __restrict__ w,
                         const float* __restrict__ b,
                         float* __restrict__ out, int rows) {
    int i = blockIdx.x * blockDim.x + threadIdx.x;
    if (i >= rows) return;
    float acc = b[0];
    const float* yr = Y + (long long)i * DD;
#pragma unroll
    for (int d = 0; d < DD; ++d) acc += yr[d] * w[d];
    out[i] = acc;
}

extern "C" void kernel_launch(void* const* d_in, const int* in_sizes, int n_in,
                              void* d_out, int out_size, void* d_ws, size_t ws_size,
                              hipStream_t stream) {
    const float* x        = (const float*)d_in[0];
    const int*   ei       = (const int*)d_in[1];
    const float* ea       = (const float*)d_in[2];
    const int*   batch    = (const int*)d_in[3];
    const float* fc1_w    = (const float*)d_in[4];
    const float* fc1_b    = (const float*)d_in[5];
    const float* fc2_w    = (const float*)d_in[6];
    const float* fc2_b    = (const float*)d_in[7];
    const float* rfc1_w   = (const float*)d_in[8];
    const float* rfc1_b   = (const float*)d_in[9];
    const float* rfc2_w   = (const float*)d_in[10];
    const float* rfc2_b   = (const float*)d_in[11];
    const float* fcout_w  = (const float*)d_in[12];
    const float* fcout_b  = (const float*)d_in[13];
    const float* bn1_g    = (const float*)d_in[14];
    const float* bn1_b    = (const float*)d_in[15];
    const float* bn2_g    = (const float*)d_in[16];
    const float* bn2_b    = (const float*)d_in[17];
    const float* rbn1_g   = (const float*)d_in[18];
    const float* rbn1_b   = (const float*)d_in[19];
    const float* rbn2_g   = (const float*)d_in[20];
    const float* rbn2_b   = (const float*)d_in[21];
    float* out = (float*)d_out;

    // workspace layout (floats): h | s | cnt | stats(4*32) | g | a1 | a2 | a3 | a4
    float* ws   = (float*)d_ws;
    float* h    = ws;
    float* s    = h + (long long)NN * DD;
    float* cnt  = s + (long long)GG * DD;
    float* stat = cnt + GG;                 // layer L: sum = stat+L*32, sq = stat+L*32+16
    float* g    = stat + 128;
    float* a1   = g  + (long long)GG * DD;
    float* a2   = a1 + (long long)GG * DD;
    float* a3   = a2 + (long long)GG * DD;
    float* a4   = a3 + (long long)GG * DD;

    const long long nzero = (long long)NN * DD + (long long)GG * DD + GG + 128;
    zero_kernel<<<2048, 256, 0, stream>>>(ws, nzero);

    edge_scatter<<<(EE + 255) / 256, 256, 0, stream>>>(x, ei, ea, h);
    node_pool<<<(NN + 255) / 256, 256, 0, stream>>>(h, batch, s, cnt);
    pool_div<<<(GG * DD + 255) / 256, 256, 0, stream>>>(s, cnt, g);

    const int ntiles = (GG + 15) / 16;              // 313
    const int gblocks = (ntiles + 7) / 8;           // 8 waves (tiles) per block
    const int bnblocks = (GG * DD + 255) / 256;

    // layer 1
    gemm_wmma<<<gblocks, 256, 0, stream>>>(g, fc1_w, fc1_b, a1, stat + 0, stat + 16, GG);
    bn_apply<<<bnblocks, 256, 0, stream>>>(a1, stat + 0, stat + 16, bn1_g, bn1_b, nullptr, GG, 0);
    // layer 2  (a2 is the residual)
    gemm_wmma<<<gblocks, 256, 0, stream>>>(a1, fc2_w, fc2_b, a2, stat + 32, stat + 48, GG);
    bn_apply<<<bnblocks, 256, 0, stream>>>(a2, stat + 32, stat + 48, bn2_g, bn2_b, nullptr, GG, 0);
    // layer 3
    gemm_wmma<<<gblocks, 256, 0, stream>>>(a2, rfc1_w, rfc1_b, a3, stat + 64, stat + 80, GG);
    bn_apply<<<bnblocks, 256, 0, stream>>>(a3, stat + 64, stat + 80, rbn1_g, rbn1_b, nullptr, GG, 0);
    // layer 4 (+ residual a2, relu after add)
    gemm_wmma<<<gblocks, 256, 0, stream>>>(a3, rfc2_w, rfc2_b, a4, stat + 96, stat + 112, GG);
    bn_apply<<<bnblocks, 256, 0, stream>>>(a4, stat + 96, stat + 112, rbn2_g, rbn2_b, a2, GG, 1);

    out_proj<<<(GG + 255) / 256, 256, 0, stream>>>(a4, fcout_w, fcout_b, out, GG);
}